// GCNLayer_47201690583744
// MI455X (gfx1250) — compile-verified
//
#include <hip/hip_runtime.h>

// ---------------------------------------------------------------------------
// GCN layer: out = D^-1/2 (A+I) D^-1/2 (X @ W)
//   X: [N,128] f32, edge_index: [2,E] i64, W: [128,128] f32, out: [N,128] f32
// ---------------------------------------------------------------------------

typedef __attribute__((ext_vector_type(2))) float v2f;
typedef __attribute__((ext_vector_type(8))) float v8f;

#define IN_DIM  128
#define OUT_DIM 128

// ---- deg init: self-loop contributes 1 to every node's degree --------------
__global__ void init_deg_kernel(float* __restrict__ deg, int n) {
    int i = blockIdx.x * blockDim.x + threadIdx.x;
    if (i < n) deg[i] = 1.0f;
}

// ---- degree count over edge rows (atomic f32 add -> global_atomic_add_f32) -
__global__ void deg_count_kernel(const long long* __restrict__ ei,
                                 float* __restrict__ deg, int E) {
    int e = blockIdx.x * blockDim.x + threadIdx.x;
    if (e < E) {
        int row = (int)ei[e];
        unsafeAtomicAdd(&deg[row], 1.0f);
    }
}

// ---- in-place deg -> deg^-1/2 (deg >= 1 always, so no zero guard needed) ---
__global__ void rsqrt_kernel(float* __restrict__ deg, int n) {
    int i = blockIdx.x * blockDim.x + threadIdx.x;
    if (i < n) deg[i] = __frsqrt_rn(deg[i]);
}

// ---- xW = X @ W via V_WMMA_F32_16X16X4_F32 ---------------------------------
// Block: 256 threads = 8 waves. Wave w computes the 16x16 tile at
// (rows blockIdx.x*16 .. +15, cols w*16 .. +15). K loop: 128 / 4 = 32 WMMAs.
//
// f32 WMMA VGPR layouts (ISA 7.12.2):
//   A (16x4):  lane L holds row M = L&15; lane-half (L>>4) selects K base
//              {0,2}; the 2 VGPRs hold K base+0, base+1.
//   B (4x16):  lane L holds col N = L&15; lane-half selects K base {0,2};
//              the 2 VGPRs hold K base+0, base+1.
//   C/D:       VGPR r, lanes 0-15 -> (M=r, N=lane); lanes 16-31 -> (M=r+8).
__global__ void gemm_xw_kernel(const float* __restrict__ X,
                               const float* __restrict__ W,
                               float* __restrict__ xW, int nrows) {
    const int lane = threadIdx.x & 31;
    const int wave = threadIdx.x >> 5;
    const int row0 = blockIdx.x * 16;
    const int n0   = wave * 16;
    const int m    = lane & 15;      // row (A) / col (B) within tile
    const int half = lane >> 4;      // K-base selector: 0 -> K{0,1}, 1 -> K{2,3}

    int arow = row0 + m;
    if (arow >= nrows) arow = nrows - 1;          // clamp (N % 16 == 0 anyway)

    const float* __restrict__ a_ptr = X + (size_t)arow * IN_DIM + half * 2;
    const float* __restrict__ b_ptr = W + (size_t)(half * 2) * OUT_DIM + n0 + m;

    v8f c = {};
#pragma unroll 8
    for (int k = 0; k < IN_DIM; k += 4) {
        v2f a = *(const v2f*)(a_ptr + k);          // A: K = k + half*2 + {0,1}
        v2f b;
        b.x = b_ptr[(size_t)k * OUT_DIM];          // B: K = k + half*2
        b.y = b_ptr[(size_t)k * OUT_DIM + OUT_DIM];//    K = k + half*2 + 1
        // 8 args: (neg_a, A, neg_b, B, c_mod, C, reuse_a, reuse_b)
        c = __builtin_amdgcn_wmma_f32_16x16x4_f32(
                false, a, false, b, (short)0, c, false, false);
    }

    // Epilogue: uniform branch -> clean unguarded stores for full tiles
    // (avoids 8x v_cmpx/s_cbranch exec-mask blocks in the common path).
    float* __restrict__ o = xW + (size_t)(row0 + half * 8) * OUT_DIM + n0 + m;
    if (row0 + 16 <= nrows) {
#pragma unroll
        for (int r = 0; r < 8; ++r)
            o[(size_t)r * OUT_DIM] = c[r];
    } else {
#pragma unroll
        for (int r = 0; r < 8; ++r)
            if (row0 + half * 8 + r < nrows)
                o[(size_t)r * OUT_DIM] = c[r];
    }
}

// ---- edge scatter: out[row] += dinv[row]*dinv[col] * xW[col] ---------------
// One wave per edge; each lane handles 4 consecutive features (float4 gather,
// 4x global_atomic_add_f32 scatter). xW and out both fit in the 192MB L2,
// so gathers + atomics stay cache-resident.
__global__ void edge_scatter_kernel(const long long* __restrict__ ei,
                                    const float* __restrict__ xW,
                                    const float* __restrict__ dinv,
                                    float* __restrict__ out, int E) {
    int e = blockIdx.x * (blockDim.x >> 5) + (threadIdx.x >> 5);
    if (e >= E) return;
    const int lane = threadIdx.x & 31;

    const int row = (int)ei[e];
    const int col = (int)ei[(size_t)E + e];
    const float norm = dinv[row] * dinv[col];

    const float4 v = *(const float4*)(xW + (size_t)col * OUT_DIM + lane * 4);
    float* __restrict__ o = out + (size_t)row * OUT_DIM + lane * 4;
    unsafeAtomicAdd(o + 0, norm * v.x);
    unsafeAtomicAdd(o + 1, norm * v.y);
    unsafeAtomicAdd(o + 2, norm * v.z);
    unsafeAtomicAdd(o + 3, norm * v.w);
}

// ---- self loop: out[i] += dinv[i]^2 * xW[i]  (runs after scatter, no atomics)
__global__ void self_loop_kernel(const float* __restrict__ xW,
                                 const float* __restrict__ dinv,
                                 float* __restrict__ out, int total) {
    int idx = blockIdx.x * blockDim.x + threadIdx.x;
    if (idx >= total) return;
    const int i = idx >> 7;            // / OUT_DIM
    const float di = dinv[i];
    out[idx] += di * di * xW[idx];
}

// ---------------------------------------------------------------------------
extern "C" void kernel_launch(void* const* d_in, const int* in_sizes, int n_in,
                              void* d_out, int out_size, void* d_ws, size_t ws_size,
                              hipStream_t stream) {
    const float*     X  = (const float*)d_in[0];       // [N, 128]
    const long long* ei = (const long long*)d_in[1];   // [2, E] int64
    const float*     W  = (const float*)d_in[2];       // [128, 128]
    float* out = (float*)d_out;

    const int N = in_sizes[0] / IN_DIM;   // 100000
    const int E = in_sizes[1] / 2;        // 1600000

    // workspace layout: xW [N*128] floats, then deg/dinv [N] floats
    float* xW  = (float*)d_ws;
    float* deg = xW + (size_t)N * OUT_DIM;

    // zero the output (atomics accumulate into it every call)
    hipMemsetAsync(out, 0, (size_t)N * OUT_DIM * sizeof(float), stream);

    // degree (init 1.0 for self loop) -> atomic count -> rsqrt in place
    init_deg_kernel<<<(N + 255) / 256, 256, 0, stream>>>(deg, N);
    deg_count_kernel<<<(E + 255) / 256, 256, 0, stream>>>(ei, deg, E);
    rsqrt_kernel<<<(N + 255) / 256, 256, 0, stream>>>(deg, N);

    // xW = X @ W  (WMMA f32 16x16x4; 8 waves/block cover full 128-col strip)
    gemm_xw_kernel<<<(N + 15) / 16, 256, 0, stream>>>(X, W, xW, N);

    // edge-wise SpMM scatter (one wave per edge), then self-loop term
    edge_scatter_kernel<<<(E + 7) / 8, 256, 0, stream>>>(ei, xW, deg, out, E);
    self_loop_kernel<<<((size_t)N * OUT_DIM + 255) / 256, 256, 0, stream>>>(
        xW, deg, out, (int)((size_t)N * OUT_DIM));
}